// MLPDecoder_1580547968774
// MI455X (gfx1250) — compile-verified
//
#include <hip/hip_runtime.h>
#include <hip/hip_bf16.h>

// out[e][c] = sum_k |input[r[e]][k] - input[c[e]][k]| * weight[k][c]
// E = 500000, D = 256, C = 4. fp32 throughout via V_WMMA_F32_16X16X4_F32.

typedef __attribute__((ext_vector_type(2))) float v2f;
typedef __attribute__((ext_vector_type(8))) float v8f;

#define DIM        256
#define NCLASS     4
#define WAVES_PB   8          // 256 threads = 8 wave32 waves
#define TILE_M     16         // edges per wave (WMMA M)

// Main kernel: processes FULL 16-edge tiles only (no per-store bounds checks,
// so the output path is 1 predicate + 8 straight global_store_b32).
__global__ __launch_bounds__(256) void
mlp_decoder_wmma_kernel(const float* __restrict__ input,
                        const int*   __restrict__ r_idx,
                        const int*   __restrict__ c_idx,
                        const float* __restrict__ weight,
                        float*       __restrict__ out,
                        int n_tiles)
{
    // Zero-padded, transposed weight staged in LDS: wlds[n][k], n in [0,16).
    // Columns n>=4 are zero; their WMMA outputs are discarded.
    __shared__ float wlds[16 * DIM];   // 16 KB of the 320 KB WGP LDS

    for (int i = threadIdx.x; i < 16 * DIM; i += 256) {
        const int n = i >> 8;          // column (class)
        const int k = i & (DIM - 1);   // dim
        wlds[i] = (n < NCLASS) ? weight[k * NCLASS + n] : 0.0f;
    }
    __syncthreads();

    const int lane = threadIdx.x & 31;
    const int wave = threadIdx.x >> 5;
    const int tile = blockIdx.x * WAVES_PB + wave;
    if (tile >= n_tiles) return;       // wave-uniform exit: EXEC stays all-1s

    const int m = lane & 15;           // WMMA A row (edge in tile) / B column
    const int h = lane >> 4;           // lane-half selects K pair {2h, 2h+1}

    const int e = tile * TILE_M + m;   // always in-bounds: full tiles only
    const float* __restrict__ rowR = input + (size_t)r_idx[e] * DIM;
    const float* __restrict__ rowC = input + (size_t)c_idx[e] * DIM;
    const float* __restrict__ wrow = &wlds[m * DIM];

    v8f acc = {0.f, 0.f, 0.f, 0.f, 0.f, 0.f, 0.f, 0.f};

    // K loop: 4 dims per WMMA, 64 WMMAs per 16-edge tile. Unroll 8 keeps up
    // to 16 b64 gathers in flight to hide random-access L2 latency.
    #pragma unroll 8
    for (int kb = 0; kb < DIM; kb += 4) {
        const int kk = kb + 2 * h;     // this lane's two contiguous K values

        const float2 xr = *(const float2*)(rowR + kk);   // global_load_b64
        const float2 xc = *(const float2*)(rowC + kk);   // global_load_b64
        const float2 bw = *(const float2*)(wrow + kk);   // ds_load_b64

        v2f a2;                         // A: M=m, K=kk..kk+1
        a2.x = __builtin_fabsf(xr.x - xc.x);
        a2.y = __builtin_fabsf(xr.y - xc.y);
        v2f b2;                         // B: N=m, K=kk..kk+1
        b2.x = bw.x;
        b2.y = bw.y;

        // D = A(16x4) * B(4x16) + C, fp32 accumulate
        acc = __builtin_amdgcn_wmma_f32_16x16x4_f32(
                  /*neg_a=*/false, a2,
                  /*neg_b=*/false, b2,
                  /*c_mod=*/(short)0, acc,
                  /*reuse_a=*/false, /*reuse_b=*/false);
    }

    // C/D layout: VGPR r holds row (r + 8*h), column n = lane&15.
    // Single EXEC predicate, then 8 unconditional stores.
    if (m < NCLASS) {
        float* __restrict__ outp =
            out + (size_t)(tile * TILE_M + 8 * h) * NCLASS + m;
        #pragma unroll
        for (int r = 0; r < 8; ++r)
            outp[(size_t)r * NCLASS] = acc[r];
    }
}

// Scalar cleanup for any edges beyond the last full 16-edge tile
// (not hit for E = 500000, but keeps the launcher fully general).
__global__ __launch_bounds__(64) void
mlp_decoder_tail_kernel(const float* __restrict__ input,
                        const int*   __restrict__ r_idx,
                        const int*   __restrict__ c_idx,
                        const float* __restrict__ weight,
                        float*       __restrict__ out,
                        int start, int n_edges)
{
    const int e = start + blockIdx.x * blockDim.x + threadIdx.x;
    if (e >= n_edges) return;
    const float* a = input + (size_t)r_idx[e] * DIM;
    const float* b = input + (size_t)c_idx[e] * DIM;
    float acc0 = 0.f, acc1 = 0.f, acc2 = 0.f, acc3 = 0.f;
    for (int k = 0; k < DIM; ++k) {
        const float d = __builtin_fabsf(a[k] - b[k]);
        acc0 += d * weight[k * NCLASS + 0];
        acc1 += d * weight[k * NCLASS + 1];
        acc2 += d * weight[k * NCLASS + 2];
        acc3 += d * weight[k * NCLASS + 3];
    }
    out[(size_t)e * NCLASS + 0] = acc0;
    out[(size_t)e * NCLASS + 1] = acc1;
    out[(size_t)e * NCLASS + 2] = acc2;
    out[(size_t)e * NCLASS + 3] = acc3;
}

extern "C" void kernel_launch(void* const* d_in, const int* in_sizes, int n_in,
                              void* d_out, int out_size, void* d_ws, size_t ws_size,
                              hipStream_t stream)
{
    const float* input  = (const float*)d_in[0];
    const int*   r_idx  = (const int*)  d_in[1];
    const int*   c_idx  = (const int*)  d_in[2];
    const float* weight = (const float*)d_in[3];
    float*       out    = (float*)d_out;

    const int n_edges = in_sizes[1];              // 500000
    const int n_tiles = n_edges / TILE_M;         // full tiles: 31250
    const int rem     = n_edges - n_tiles * TILE_M;

    if (n_tiles > 0) {
        const int blocks = (n_tiles + WAVES_PB - 1) / WAVES_PB;
        mlp_decoder_wmma_kernel<<<blocks, 256, 0, stream>>>(
            input, r_idx, c_idx, weight, out, n_tiles);
    }
    if (rem > 0) {
        mlp_decoder_tail_kernel<<<(rem + 63) / 64, 64, 0, stream>>>(
            input, r_idx, c_idx, weight, out, n_tiles * TILE_M, n_edges);
    }
}